// Fusion_1640677507707
// MI455X (gfx1250) — compile-verified
//
#include <hip/hip_runtime.h>
#include <math.h>

// CDNA5 / gfx1250: wave32, WMMA f32 16x16x4.
typedef __attribute__((ext_vector_type(2))) float v2f;
typedef __attribute__((ext_vector_type(8))) float v8f;

#define DIM 16
#define KNB 16

// One wave = 32 nodes. Block = 256 threads = 8 waves = 256 nodes.
// Phase 1 (lane = node): gather 16 neighbor rows, online softmax over
//   logits = nbr . Wa[16:32] (self term + bias cancel in softmax),
//   accumulate weighted neighbor vector v[16]. Stage v in LDS.
// Phase 2 (wave): two 16-node tiles; out_tile = V(16x16) @ Wf^T via
//   4 chained V_WMMA_F32_16X16X4_F32 per tile.
__global__ __launch_bounds__(256) void fusion_kernel(
    const float* __restrict__ self_emb,   // [N,16]   (read only when add_self)
    const float* __restrict__ nbr_tab,    // [M,16]
    const int*   __restrict__ idx,        // [N,16]
    const float* __restrict__ Wf,         // [16,16]  row-major; fo = nbr @ Wf^T
    const float* __restrict__ Wa,         // [32]     only Wa[16..31] matters
    float*       __restrict__ out,        // [N,16]
    int N, int add_self)
{
    // per-wave staging: 32 nodes x 16 dims, stride 17 floats (bank-conflict-free)
    __shared__ float ldsv[8][32 * 17];

    const int lane     = threadIdx.x & 31;
    const int wave     = threadIdx.x >> 5;
    const int waveBase = blockIdx.x * 256 + wave * 32;  // first node of this wave
    const int node     = waveBase + lane;
    const int nodeC    = node < N ? node : (N - 1);     // clamp: keep EXEC full

    const int half = lane >> 4;   // 0: lanes 0-15, 1: lanes 16-31
    const int lm   = lane & 15;   // A: row m ; B/C/D: col e

    // ---- B chunks (issued first; long done by WMMA time) ----
    // b[i] lane holds B[4i+2*half + j][e] = Wf[e][4i+2*half + j]
    v2f bmat[4];
    #pragma unroll
    for (int i = 0; i < 4; ++i) {
        const float* wp = Wf + lm * DIM + 4 * i + 2 * half;
        v2f b; b.x = wp[0]; b.y = wp[1];
        bmat[i] = b;
    }

    // ---- attention weights for neighbor half (uniform -> scalarized loads) ----
    float waN[DIM];
    #pragma unroll
    for (int d = 0; d < DIM; ++d) waN[d] = Wa[DIM + d];

    // ---- neighbor indices for this lane's node ----
    int nb[KNB];
    {
        const int4* ip = reinterpret_cast<const int4*>(idx + (size_t)nodeC * KNB);
        int4 a0 = ip[0], a1 = ip[1], a2 = ip[2], a3 = ip[3];
        nb[0]=a0.x;  nb[1]=a0.y;  nb[2]=a0.z;  nb[3]=a0.w;
        nb[4]=a1.x;  nb[5]=a1.y;  nb[6]=a1.z;  nb[7]=a1.w;
        nb[8]=a2.x;  nb[9]=a2.y;  nb[10]=a2.z; nb[11]=a2.w;
        nb[12]=a3.x; nb[13]=a3.y; nb[14]=a3.z; nb[15]=a3.w;
    }

    // ---- single-pass gather + online softmax + weighted sum ----
    float m = -INFINITY, s = 0.0f;
    float v[DIM];
    #pragma unroll
    for (int d = 0; d < DIM; ++d) v[d] = 0.0f;

    #pragma unroll
    for (int k = 0; k < KNB; ++k) {
        const float4* rp = reinterpret_cast<const float4*>(nbr_tab + (size_t)nb[k] * DIM);
        float4 r0 = rp[0], r1 = rp[1], r2 = rp[2], r3 = rp[3];
        float r[DIM] = { r0.x, r0.y, r0.z, r0.w,
                         r1.x, r1.y, r1.z, r1.w,
                         r2.x, r2.y, r2.z, r2.w,
                         r3.x, r3.y, r3.z, r3.w };
        float l = 0.0f;
        #pragma unroll
        for (int d = 0; d < DIM; ++d) l = fmaf(r[d], waN[d], l);

        float mn   = fmaxf(m, l);
        float corr = __expf(m - mn);   // first iter: exp(-inf) == 0
        float p    = __expf(l - mn);
        s = s * corr + p;
        #pragma unroll
        for (int d = 0; d < DIM; ++d) v[d] = v[d] * corr + p * r[d];
        m = mn;
    }
    const float inv = 1.0f / s;
    #pragma unroll
    for (int d = 0; d < DIM; ++d) v[d] *= inv;

    // ---- issue accumulator-base loads NOW (clamped, EXEC stays full);
    //      they overlap the barrier + LDS + WMMA below ----
    const float* __restrict__ srcBase = add_self ? self_emb : out;
    float baseAcc[2][8];
    #pragma unroll
    for (int t = 0; t < 2; ++t) {
        #pragma unroll
        for (int r = 0; r < 8; ++r) {
            int g  = waveBase + t * 16 + 8 * half + r;
            int gc = g < N ? g : (N - 1);
            baseAcc[t][r] = srcBase[(size_t)gc * DIM + lm];
        }
    }

    // ---- stage v into LDS in plain per-node layout ----
    {
        float* vd = &ldsv[wave][lane * 17];
        #pragma unroll
        for (int d = 0; d < DIM; ++d) vd[d] = v[d];
    }
    __syncthreads();

    // ---- WMMA: out[m,e] = sum_d v[m,d] * Wf[e,d]  (A = v tile, B = Wf^T) ----
    const int fullTile = (waveBase + 32 <= N);   // wave-uniform

    #pragma unroll
    for (int t = 0; t < 2; ++t) {
        v8f c = {};
        #pragma unroll
        for (int i = 0; i < 4; ++i) {
            // A chunk: lane holds v[node = t*16+lm][4i+2*half + j]
            const float* ap = &ldsv[wave][(t * 16 + lm) * 17 + 4 * i + 2 * half];
            v2f a; a.x = ap[0]; a.y = ap[1];
            c = __builtin_amdgcn_wmma_f32_16x16x4_f32(
                    false, a, false, bmat[i], (short)0, c, false, false);
        }
        // C/D layout: lane -> col e = lm; VGPR r -> row m = r + 8*half
        float* op = out + (size_t)(waveBase + t * 16 + 8 * half) * DIM + lm;
        if (fullTile) {
            #pragma unroll
            for (int r = 0; r < 8; ++r)
                op[(size_t)r * DIM] = baseAcc[t][r] + c[r];
        } else {
            #pragma unroll
            for (int r = 0; r < 8; ++r) {
                int g = waveBase + t * 16 + 8 * half + r;
                if (g < N)
                    op[(size_t)r * DIM] = baseAcc[t][r] + c[r];
            }
        }
    }
}

extern "C" void kernel_launch(void* const* d_in, const int* in_sizes, int n_in,
                              void* d_out, int out_size, void* d_ws, size_t ws_size,
                              hipStream_t stream) {
    const float* concept_emb = (const float*)d_in[0];
    const float* item_emb    = (const float*)d_in[1];
    const float* stu_emb     = (const float*)d_in[2];
    const float* c_Wf1 = (const float*)d_in[3];
    const float* c_Wf2 = (const float*)d_in[4];
    const float* c_Wa1 = (const float*)d_in[5];
    const float* c_Wa2 = (const float*)d_in[7];
    const float* i_Wf1 = (const float*)d_in[9];
    const float* i_Wf2 = (const float*)d_in[10];
    const float* i_Wa1 = (const float*)d_in[11];
    const float* i_Wa2 = (const float*)d_in[13];
    const float* s_Wf  = (const float*)d_in[15];
    const float* s_Wa  = (const float*)d_in[16];
    const int* cc_idx = (const int*)d_in[18];
    const int* ci_idx = (const int*)d_in[19];
    const int* ic_idx = (const int*)d_in[20];
    const int* is_idx = (const int*)d_in[21];
    const int* si_idx = (const int*)d_in[22];

    const int NC = in_sizes[0] / DIM;
    const int NI = in_sizes[1] / DIM;
    const int NS = in_sizes[2] / DIM;

    float* outC = (float*)d_out;
    float* outI = outC + (size_t)NC * DIM;
    float* outS = outI + (size_t)NI * DIM;

    const int T = 256;
    dim3 bC((NC + T - 1) / T), bI((NI + T - 1) / T), bS((NS + T - 1) / T);

    // concepts = self + fus(cc, concept nbrs) + fus(ci, item nbrs)
    fusion_kernel<<<bC, T, 0, stream>>>(concept_emb, concept_emb, cc_idx, c_Wf1, c_Wa1, outC, NC, 1);
    fusion_kernel<<<bC, T, 0, stream>>>(concept_emb, item_emb,    ci_idx, c_Wf2, c_Wa2, outC, NC, 0);
    // items = self + fus(ic, concept nbrs) + fus(is, stu nbrs)
    fusion_kernel<<<bI, T, 0, stream>>>(item_emb, concept_emb, ic_idx, i_Wf1, i_Wa1, outI, NI, 1);
    fusion_kernel<<<bI, T, 0, stream>>>(item_emb, stu_emb,     is_idx, i_Wf2, i_Wa2, outI, NI, 0);
    // stus = self + fus(si, item nbrs)
    fusion_kernel<<<bS, T, 0, stream>>>(stu_emb, item_emb, si_idx, s_Wf, s_Wa, outS, NS, 1);
}